// GaussianGenerator_30227979829381
// MI455X (gfx1250) — compile-verified
//
#include <hip/hip_runtime.h>
#include <hip/hip_bf16.h>
#include <math.h>

// ---------------------------------------------------------------------------
// GaussianGenerator: rank-1 LowRankMVN log_prob -> softmax over vocab -> x @ P
// VOCAB=100000, VEC=300, K=256, B=1024. All f32; uses V_WMMA_F32_16X16X4_F32.
// ---------------------------------------------------------------------------

typedef float v2f __attribute__((ext_vector_type(2)));
typedef float v8f __attribute__((ext_vector_type(8)));

#define VOCABN 100000
#define VECN   300
#define KTOP   256
#define BATCH  1024
#define LN2PI  1.8378770664093453f

__device__ __forceinline__ v8f wmma_f32(v2f a, v2f b, v8f c) {
  // D = A(16x4 f32) x B(4x16 f32) + C(16x16 f32)
  return __builtin_amdgcn_wmma_f32_16x16x4_f32(
      /*neg_a=*/false, a, /*neg_b=*/false, b,
      /*c_mod=*/(short)0, c, /*reuse_a=*/false, /*reuse_b=*/false);
}

// ---------------------------------------------------------------------------
// Phase A: per-topic constants + WMMA-friendly B matrices laid out [vec][k].
//   Bq1[j][k] = 1/d            (multiplies wv^2)
//   Bq2[j][k] = -2*mu/d        (multiplies wv)
//   Bu [j][k] = w/d            (multiplies wv)
//   kconst[k] = { c1 = sum(mu^2/d), c2 = sum(mu*w/d), 1/cap,
//                 base = -0.5*(V*ln2pi + sum(log d) + log cap) }
// ---------------------------------------------------------------------------
__global__ void __launch_bounds__(256)
prep_kernel(const float* __restrict__ mu, const float* __restrict__ cf,
            const float* __restrict__ cd,
            float* __restrict__ Bq1, float* __restrict__ Bq2,
            float* __restrict__ Bu, float4* __restrict__ kconst) {
  __shared__ float red[4][256];
  const int k = blockIdx.x;
  const int tid = threadIdx.x;
  float s_cap = 0.f, s_c1 = 0.f, s_c2 = 0.f, s_ld = 0.f;
  for (int j = tid; j < VECN; j += 256) {
    float w  = cf[k * VECN + j];
    float d  = cd[k * VECN + j];
    float m  = mu[k * VECN + j];
    float id = 1.0f / d;
    float aa = w * id;
    Bq1[j * KTOP + k] = id;
    Bq2[j * KTOP + k] = -2.0f * m * id;
    Bu [j * KTOP + k] = aa;
    s_cap += w * aa;
    s_c1  += m * m * id;
    s_c2  += m * aa;
    s_ld  += __logf(d);
  }
  red[0][tid] = s_cap; red[1][tid] = s_c1; red[2][tid] = s_c2; red[3][tid] = s_ld;
  __syncthreads();
  for (int off = 128; off > 0; off >>= 1) {
    if (tid < off) {
      red[0][tid] += red[0][tid + off];
      red[1][tid] += red[1][tid + off];
      red[2][tid] += red[2][tid + off];
      red[3][tid] += red[3][tid + off];
    }
    __syncthreads();
  }
  if (tid == 0) {
    float cap  = 1.0f + red[0][0];
    float base = -0.5f * ((float)VECN * LN2PI + red[3][0] + __logf(cap));
    kconst[k] = make_float4(red[1][0], red[2][0], 1.0f / cap, base);
  }
}

// ---------------------------------------------------------------------------
// Phase B: lp[k][v] via fp32 WMMA.  Wave tile = 16 vocab x 64 topics.
// Block = 8 waves stacked over vocab (128 rows) x 64 topics.
// Inner dim VEC=300 = 75 steps of 4.
// ---------------------------------------------------------------------------
__global__ void __launch_bounds__(256)
logprob_kernel(const float* __restrict__ wv,
               const float* __restrict__ Bq1, const float* __restrict__ Bq2,
               const float* __restrict__ Bu,
               const float4* __restrict__ kconst,
               float* __restrict__ lp) {
  const int tid  = threadIdx.x;
  const int lane = tid & 31;
  const int wave = tid >> 5;
  const int lo   = lane & 15;
  const int hi   = lane >> 4;

  const int vb = blockIdx.x * 128 + wave * 16;  // this wave's 16 vocab rows
  const int kb = blockIdx.y * 64;               // this block's 64 topics

  int vrow = vb + lo;
  if (vrow > VOCABN - 1) vrow = VOCABN - 1;     // clamp (EXEC must stay full)
  const float* __restrict__ arow = wv + (size_t)vrow * VECN;

  v8f accq[4] = {};
  v8f accu[4] = {};

  for (int j = 0; j < VECN; j += 4) {
    v2f a, a2;
    a.x  = arow[j + hi * 2];
    a.y  = arow[j + hi * 2 + 1];
    a2.x = a.x * a.x;
    a2.y = a.y * a.y;                  // elementwise square == wv^2 fragment
    const int r0 = (j + hi * 2) * KTOP;
    const int r1 = r0 + KTOP;
#pragma unroll
    for (int t = 0; t < 4; ++t) {
      const int kc = kb + t * 16 + lo;
      v2f b1, b2, b3;
      b1.x = Bq1[r0 + kc]; b1.y = Bq1[r1 + kc];
      b2.x = Bq2[r0 + kc]; b2.y = Bq2[r1 + kc];
      b3.x = Bu [r0 + kc]; b3.y = Bu [r1 + kc];
      accq[t] = wmma_f32(a2, b1, accq[t]);   // + wv^2 * inv_d
      accq[t] = wmma_f32(a,  b2, accq[t]);   // - 2 wv * mu*inv_d
      accu[t] = wmma_f32(a,  b3, accu[t]);   // + wv * a
    }
  }

#pragma unroll
  for (int t = 0; t < 4; ++t) {
    const int topic = kb + t * 16 + lo;
    const float4 kc = kconst[topic];
    const size_t rowbase = (size_t)topic * VOCABN;
#pragma unroll
    for (int i = 0; i < 8; ++i) {
      const int v = vb + hi * 8 + i;           // C/D row m = i + 8*(lane>>4)
      if (v < VOCABN) {
        float q    = accq[t][i] + kc.x;
        float u    = accu[t][i] - kc.y;
        float maha = q - u * u * kc.z;
        lp[rowbase + v] = kc.w - 0.5f * maha;
      }
    }
  }
}

// ---------------------------------------------------------------------------
// Phase C: per-topic softmax stats over vocab (max + sum-exp).
// ---------------------------------------------------------------------------
__global__ void __launch_bounds__(256)
softmax_stats_kernel(const float* __restrict__ lp,
                     float* __restrict__ mtab, float* __restrict__ invZ) {
  __shared__ float red[256];
  const int k = blockIdx.x;
  const int tid = threadIdx.x;
  const float* __restrict__ row = lp + (size_t)k * VOCABN;

  float m = -3.4e38f;
  for (int v = tid; v < VOCABN; v += 256) m = fmaxf(m, row[v]);
  red[tid] = m; __syncthreads();
  for (int off = 128; off > 0; off >>= 1) {
    if (tid < off) red[tid] = fmaxf(red[tid], red[tid + off]);
    __syncthreads();
  }
  m = red[0];
  __syncthreads();

  float s = 0.f;
  for (int v = tid; v < VOCABN; v += 256) s += __expf(row[v] - m);
  red[tid] = s; __syncthreads();
  for (int off = 128; off > 0; off >>= 1) {
    if (tid < off) red[tid] += red[tid + off];
    __syncthreads();
  }
  if (tid == 0) { mtab[k] = m; invZ[k] = 1.0f / red[0]; }
}

// ---------------------------------------------------------------------------
// x'[b,k] = x[b,k] / Z_k
// ---------------------------------------------------------------------------
__global__ void __launch_bounds__(256)
scale_x_kernel(const float* __restrict__ x, const float* __restrict__ invZ,
               float* __restrict__ xp) {
  const int i = blockIdx.x * 256 + threadIdx.x;
  xp[i] = x[i] * invZ[i & (KTOP - 1)];
}

// ---------------------------------------------------------------------------
// Phase D: out(1024 x 100000) = x'(1024x256) @ exp(lp - m).
// Block tile 32(M) x 512(N); 8 waves of 32x64; inner K = 256 (64 WMMA steps).
// A panel staged in LDS with padded stride (260) to avoid bank conflicts;
// exp(lp-m) B-fragments computed on the fly (lp-m <= 0, so exp in (0,1]).
// ---------------------------------------------------------------------------
#define ASTRIDE 260
__global__ void __launch_bounds__(256)
out_gemm_kernel(const float* __restrict__ xp, const float* __restrict__ lp,
                const float* __restrict__ mtab, float* __restrict__ out) {
  __shared__ float As[32 * ASTRIDE];
  __shared__ float Ms[KTOP];
  const int tid  = threadIdx.x;
  const int lane = tid & 31;
  const int wave = tid >> 5;
  const int lo   = lane & 15;
  const int hi   = lane >> 4;

  const int mb = blockIdx.y * 32;
  const int nb = blockIdx.x * 512 + wave * 64;

  // Stage x' block panel (32 x 256) and the per-topic max table.
  for (int idx = tid; idx < 32 * KTOP; idx += 256) {
    const int r = idx >> 8, c = idx & 255;
    As[r * ASTRIDE + c] = xp[(size_t)(mb + r) * KTOP + c];
  }
  Ms[tid] = mtab[tid];
  __syncthreads();

  int ncol[4];
#pragma unroll
  for (int jt = 0; jt < 4; ++jt) {
    int n = nb + jt * 16 + lo;
    ncol[jt] = (n < VOCABN) ? n : (VOCABN - 1);   // clamp loads; stores guarded
  }

  v8f acc[2][4] = {};

  for (int k0 = 0; k0 < KTOP; k0 += 4) {
    const int kr = k0 + hi * 2;
    v2f a0, a1;
    a0.x = As[lo * ASTRIDE + kr];
    a0.y = As[lo * ASTRIDE + kr + 1];
    a1.x = As[(lo + 16) * ASTRIDE + kr];
    a1.y = As[(lo + 16) * ASTRIDE + kr + 1];
    const float m0 = Ms[kr];
    const float m1 = Ms[kr + 1];
    const size_t r0 = (size_t)kr * VOCABN;
    const size_t r1 = r0 + VOCABN;
#pragma unroll
    for (int jt = 0; jt < 4; ++jt) {
      v2f b;
      b.x = __expf(lp[r0 + ncol[jt]] - m0);
      b.y = __expf(lp[r1 + ncol[jt]] - m1);
      acc[0][jt] = wmma_f32(a0, b, acc[0][jt]);
      acc[1][jt] = wmma_f32(a1, b, acc[1][jt]);
    }
  }

#pragma unroll
  for (int it = 0; it < 2; ++it) {
#pragma unroll
    for (int jt = 0; jt < 4; ++jt) {
      const int n = nb + jt * 16 + lo;
      if (n < VOCABN) {
#pragma unroll
        for (int i = 0; i < 8; ++i) {
          const int m = mb + it * 16 + hi * 8 + i;
          out[(size_t)m * VOCABN + n] = acc[it][jt][i];
        }
      }
    }
  }
}

// ---------------------------------------------------------------------------
extern "C" void kernel_launch(void* const* d_in, const int* in_sizes, int n_in,
                              void* d_out, int out_size, void* d_ws, size_t ws_size,
                              hipStream_t stream) {
  const float* x  = (const float*)d_in[0];   // (1024, 256)
  const float* wv = (const float*)d_in[1];   // (100000, 300)
  const float* mu = (const float*)d_in[2];   // (256, 300)
  const float* cf = (const float*)d_in[3];   // (256, 300, 1)
  const float* cd = (const float*)d_in[4];   // (256, 300)
  float* out = (float*)d_out;                // (1024, 100000)

  char* ws = (char*)d_ws;
  size_t off = 0;
  auto carve = [&](size_t bytes) -> void* {
    void* p = ws + off;
    off += (bytes + 255) & ~(size_t)255;
    return p;
  };
  float*  Bq1    = (float*) carve((size_t)VECN * KTOP * 4);
  float*  Bq2    = (float*) carve((size_t)VECN * KTOP * 4);
  float*  Bu     = (float*) carve((size_t)VECN * KTOP * 4);
  float4* kconst = (float4*)carve((size_t)KTOP * 16);
  float*  mtab   = (float*) carve((size_t)KTOP * 4);
  float*  invZ   = (float*) carve((size_t)KTOP * 4);
  float*  xprime = (float*) carve((size_t)BATCH * KTOP * 4);
  float*  lp     = (float*) carve((size_t)KTOP * VOCABN * 4);
  (void)ws_size; (void)in_sizes; (void)n_in; (void)out_size;

  prep_kernel<<<KTOP, 256, 0, stream>>>(mu, cf, cd, Bq1, Bq2, Bu, kconst);

  dim3 g1((VOCABN + 127) / 128, KTOP / 64);
  logprob_kernel<<<g1, 256, 0, stream>>>(wv, Bq1, Bq2, Bu, kconst, lp);

  softmax_stats_kernel<<<KTOP, 256, 0, stream>>>(lp, mtab, invZ);

  scale_x_kernel<<<(BATCH * KTOP) / 256, 256, 0, stream>>>(x, invZ, xprime);

  dim3 g3((VOCABN + 511) / 512, BATCH / 32);
  out_gemm_kernel<<<g3, 256, 0, stream>>>(xprime, lp, mtab, out);
}